// SupJSD_592705487350
// MI455X (gfx1250) — compile-verified
//
#include <hip/hip_runtime.h>
#include <hip/hip_bf16.h>

// SupJSD on MI455X (gfx1250, wave32).
//
// out = (1/C) * [ sum_i H_i / counts[lab_i]  -  sum_{k,c} mix*log(max(mix,eps)) ]
// with H_i = S_i/Z_i - log Z_i  (S = sum (x-m)e^{x-m}, Z = sum e^{x-m}).
// One HBM pass over 196.6MB of logits; mix accumulator (4MB) stays in L2.
// Wave reductions use V_WMMA_F32_16X16X4_F32 (B = ones -> row sums).

#define C_DIM 1000
#define NCLS  1000
#define BPV   16384     // rows per view
#define BLK   256
#define EPSF  1e-7f

typedef __attribute__((ext_vector_type(2))) float v2f;
typedef __attribute__((ext_vector_type(8))) float v8f;

// Full-wave (32-lane) FP32 sum using the matrix pipe:
// A(16x4) holds each lane's value once (other slot 0), B(4x16) = ones,
// so D[m][n] = rowsum_m; sum of the 8 D VGPRs gives half-wave sums
// (lanes 0-15: rows 0-7, lanes 16-31: rows 8-15); one shfl_xor(16) completes.
__device__ __forceinline__ float wave_sum_wmma(float v) {
    v2f a; a[0] = v;   a[1] = 0.0f;
    v2f b; b[0] = 1.0f; b[1] = 1.0f;
    v8f c = {0.f, 0.f, 0.f, 0.f, 0.f, 0.f, 0.f, 0.f};
    v8f d = __builtin_amdgcn_wmma_f32_16x16x4_f32(
        false, a, false, b, (short)0, c, false, false);
    float s = ((d[0] + d[1]) + (d[2] + d[3])) + ((d[4] + d[5]) + (d[6] + d[7]));
    s += __shfl_xor(s, 16, 32);
    return s;   // every lane holds the full 32-lane sum
}

__global__ void __launch_bounds__(BLK) zero_ws_kernel(unsigned int* ws, int nwords) {
    int idx = blockIdx.x * BLK + threadIdx.x;
    if (idx < nwords) ws[idx] = 0u;
}

__global__ void __launch_bounds__(BLK) hist_kernel(const int* __restrict__ labels,
                                                   int* __restrict__ counts) {
    int i = blockIdx.x * BLK + threadIdx.x;
    if (i < BPV) atomicAdd(&counts[labels[i]], 3);   // labels tiled 3x
}

__global__ void __launch_bounds__(BLK) row_pass_kernel(
    const float* __restrict__ lc, const float* __restrict__ l1,
    const float* __restrict__ l2, const int* __restrict__ labels,
    const int* __restrict__ counts, float* __restrict__ mixsum,
    float* __restrict__ partials)
{
    __shared__ float sx[C_DIM];
    __shared__ float red[16];
    __shared__ float bc[3];

    const int tid  = threadIdx.x;
    const int row  = blockIdx.x;           // 0 .. 3*BPV-1
    const int view = row >> 14;            // row / 16384
    const int i    = row & (BPV - 1);
    const float* src = (view == 0 ? lc : (view == 1 ? l1 : l2)) + (size_t)i * C_DIM;
    const int lab = labels[i];

    // Stage the row through LDS (single coalesced HBM read).
    for (int c = tid; c < C_DIM; c += BLK) sx[c] = src[c];
    __syncthreads();

    // ---- row max (shuffle + LDS) ----
    float m = -3.402823e38f;
    for (int c = tid; c < C_DIM; c += BLK) m = fmaxf(m, sx[c]);
    #pragma unroll
    for (int off = 16; off >= 1; off >>= 1) m = fmaxf(m, __shfl_xor(m, off, 32));
    if ((tid & 31) == 0) red[tid >> 5] = m;
    __syncthreads();
    if (tid == 0) {
        float mm = red[0];
        #pragma unroll
        for (int w = 1; w < 8; ++w) mm = fmaxf(mm, red[w]);
        bc[0] = mm;
    }
    __syncthreads();
    m = bc[0];

    // ---- Z = sum exp(x-m), S = sum (x-m)*exp(x-m)  (WMMA wave reduce) ----
    float z = 0.f, s = 0.f;
    for (int c = tid; c < C_DIM; c += BLK) {
        float t = sx[c] - m;
        float e = __expf(t);
        z += e;
        s = __fmaf_rn(t, e, s);
    }
    float zw = wave_sum_wmma(z);
    float sw = wave_sum_wmma(s);
    if ((tid & 31) == 0) { red[tid >> 5] = zw; red[8 + (tid >> 5)] = sw; }
    __syncthreads();
    if (tid == 0) {
        float Z = 0.f, S = 0.f;
        #pragma unroll
        for (int w = 0; w < 8; ++w) { Z += red[w]; S += red[8 + w]; }
        bc[1] = Z; bc[2] = S;
    }
    __syncthreads();

    const float Z    = bc[1];
    const float invZ = 1.0f / Z;

    // ---- scatter probabilities into the L2-resident mix accumulator ----
    float* mrow = mixsum + (size_t)lab * C_DIM;
    for (int c = tid; c < C_DIM; c += BLK) {
        float p = __expf(sx[c] - m) * invZ;
        atomicAdd(&mrow[c], p);
    }

    if (tid == 0) {
        float H = bc[2] * invZ - __logf(Z);          // sum_c p*log p
        atomicAdd(&partials[0], H / (float)counts[lab]);
    }
}

__global__ void __launch_bounds__(BLK) mix_pass_kernel(
    const float* __restrict__ mixsum, const int* __restrict__ counts,
    float* __restrict__ partials)
{
    __shared__ float red[8];
    int idx = blockIdx.x * BLK + threadIdx.x;
    float term = 0.f;
    if (idx < NCLS * C_DIM) {
        int   k   = idx / C_DIM;
        float cnt = fmaxf((float)counts[k], 1.0f);
        float t   = mixsum[idx] / cnt;               // mix value
        term = t * __logf(fmaxf(t, EPSF));           // mix * log(clip(mix,eps))
    }
    float w = wave_sum_wmma(term);
    if ((threadIdx.x & 31) == 0) red[threadIdx.x >> 5] = w;
    __syncthreads();
    if (threadIdx.x == 0) {
        float S = 0.f;
        #pragma unroll
        for (int j = 0; j < 8; ++j) S += red[j];
        atomicAdd(&partials[1], S);
    }
}

__global__ void finalize_kernel(const float* __restrict__ partials,
                                float* __restrict__ out) {
    if (threadIdx.x == 0 && blockIdx.x == 0)
        out[0] = (partials[0] - partials[1]) * (1.0f / (float)C_DIM);
}

extern "C" void kernel_launch(void* const* d_in, const int* in_sizes, int n_in,
                              void* d_out, int out_size, void* d_ws, size_t ws_size,
                              hipStream_t stream) {
    const float* lc     = (const float*)d_in[0];
    const float* l1     = (const float*)d_in[1];
    const float* l2     = (const float*)d_in[2];
    const int*   labels = (const int*)d_in[3];   // JAX x64 disabled -> int32

    // Workspace layout (needs ~4.01 MB):
    //   [0,4000)      counts  (int[1000])
    //   [4096,4104)   partials (float[2]): [0]=sum H/cnt, [1]=sum mix*log_mix
    //   [8192, +4MB)  mixsum  (float[1000*1000])
    char*  ws       = (char*)d_ws;
    int*   counts   = (int*)ws;
    float* partials = (float*)(ws + 4096);
    float* mixsum   = (float*)(ws + 8192);

    const int nwords = 8192 / 4 + NCLS * C_DIM;   // zero everything through mixsum
    zero_ws_kernel<<<(nwords + BLK - 1) / BLK, BLK, 0, stream>>>((unsigned int*)ws, nwords);

    hist_kernel<<<(BPV + BLK - 1) / BLK, BLK, 0, stream>>>(labels, counts);

    row_pass_kernel<<<3 * BPV, BLK, 0, stream>>>(lc, l1, l2, labels, counts,
                                                 mixsum, partials);

    mix_pass_kernel<<<(NCLS * C_DIM + BLK - 1) / BLK, BLK, 0, stream>>>(
        mixsum, counts, partials);

    finalize_kernel<<<1, 1, 0, stream>>>(partials, (float*)d_out);
}